// FCOS_53051436040647
// MI455X (gfx1250) — compile-verified
//
#include <hip/hip_runtime.h>

// Greedy class-aware NMS, faithful to the JAX reference:
//   offsets = class_id * (boxes.max() + 1); greedy suppress iou > 0.5 in
//   descending-score order (stable ties by index); out = keep ? score : 0.
//
// MI455X design: NMS is min/max interval algebra -> WMMA inapplicable.
// CDNA5-specific paths used instead: async global->LDS DMA (ASYNCcnt ring
// buffer in the serial scan), LDS atomics, global_prefetch, wave32 sizing.
// Split pipeline: parallel bitmask build across all WGPs (2MB mask lives in
// the 192MB L2), then a single-wave greedy bit-scan with async row streaming.

#define N_BOX      4096
#define NTHREADS   1024
#define MASK_WORDS (N_BOX / 32)   // 128 words = 512B per row
#define RING_DEPTH 8
#define IOU_THR    0.5f

// ---------------------------------------------------------------- kernel A
__global__ __launch_bounds__(256)
void nms_maxred_kernel(const float* __restrict__ boxes, float* __restrict__ maxc) {
    __shared__ float red[256];
    int tid = threadIdx.x;
    float m = -3.402823466e38f;
    for (int i = tid; i < N_BOX * 4; i += 256) m = fmaxf(m, boxes[i]);
    red[tid] = m;
    __syncthreads();
    for (int s = 128; s > 0; s >>= 1) {
        if (tid < s) red[tid] = fmaxf(red[tid], red[tid + s]);
        __syncthreads();
    }
    if (tid == 0) *maxc = red[0];
}

// ---------------------------------------------------------------- kernel B
// LDS bitonic sort of 4096 64-bit keys (score_bits<<32 | ~idx), descending:
// == stable argsort(-scores). Emits sorted offset-boxes + original indices.
__global__ __launch_bounds__(NTHREADS)
void nms_sort_kernel(const float* __restrict__ scores,
                     const float* __restrict__ boxes,
                     const int*   __restrict__ class_ids,
                     const float* __restrict__ maxc_p,
                     float4* __restrict__ sboxes,
                     int*    __restrict__ sidx) {
    __shared__ unsigned long long key[N_BOX];
    int tid = threadIdx.x;
    for (int i = tid; i < N_BOX; i += NTHREADS) {
        unsigned sb = __float_as_uint(scores[i]);
        key[i] = ((unsigned long long)sb << 32) | (unsigned)(~i);
    }
    __syncthreads();
    for (int k = 2; k <= N_BOX; k <<= 1) {
        for (int j = k >> 1; j > 0; j >>= 1) {
            for (int e = tid; e < N_BOX; e += NTHREADS) {
                int p = e ^ j;
                if (p > e) {
                    unsigned long long a = key[e], b = key[p];
                    bool descSeg = ((e & k) == 0);
                    if (descSeg ? (a < b) : (a > b)) { key[e] = b; key[p] = a; }
                }
            }
            __syncthreads();
        }
    }
    float off_scale = *maxc_p + 1.0f;
    for (int s = tid; s < N_BOX; s += NTHREADS) {
        unsigned orig = ~(unsigned)(key[s] & 0xFFFFFFFFull);
        sidx[s] = (int)orig;
        float off = (float)class_ids[orig] * off_scale;
        float4 b;
        b.x = boxes[orig * 4 + 0] + off;
        b.y = boxes[orig * 4 + 1] + off;
        b.z = boxes[orig * 4 + 2] + off;
        b.w = boxes[orig * 4 + 3] + off;
        sboxes[s] = b;
    }
}

__device__ __forceinline__ float box_area(float4 b) {
    return fmaxf(b.z - b.x, 0.0f) * fmaxf(b.w - b.y, 0.0f);
}
__device__ __forceinline__ bool iou_gt(float4 bi, float ai, float4 bj) {
    float aj    = box_area(bj);
    float ix    = fminf(bi.z, bj.z) - fmaxf(bi.x, bj.x);
    float iy    = fminf(bi.w, bj.w) - fmaxf(bi.y, bj.y);
    float inter = fmaxf(ix, 0.0f) * fmaxf(iy, 0.0f);
    float uni   = ai + aj - inter;
    return (inter / uni) > IOU_THR;
}

// ---------------------------------------------------------------- kernel C1
// Parallel suppression-bitmask build: mask[row][cb] bit b = 1 iff
// col = cb*32+b > row and iou(row,col) > thr. 2048 blocks -> whole GPU.
__global__ __launch_bounds__(256)
void nms_mask_kernel(const float4* __restrict__ sboxes,
                     unsigned* __restrict__ mask) {
    __shared__ float4 cb[32];
    int row  = blockIdx.x * 256 + threadIdx.x;
    int cbk  = blockIdx.y;
    int col0 = cbk * 32;
    if (threadIdx.x < 32) cb[threadIdx.x] = sboxes[col0 + threadIdx.x];
    __syncthreads();
    unsigned m = 0u;
    if (col0 + 31 > row) {                 // block not entirely below diagonal
        float4 bi = sboxes[row];
        float  ai = box_area(bi);
        #pragma unroll 8
        for (int b = 0; b < 32; ++b) {
            int col = col0 + b;
            if (col > row && iou_gt(bi, ai, cb[b])) m |= (1u << b);
        }
    }
    mask[row * MASK_WORDS + cbk] = m;
}

// ---------------------------------------------------------------- kernel C2
// Single-wave greedy bit-scan. Rows streamed global->LDS through an 8-deep
// async ring (global_load_async_to_lds_b128, in-order completion), so the
// serial chain only pays ds_load + v_or per kept row. Lane l exclusively
// owns keep-mask words 4l..4l+3 (no atomics).
__global__ __launch_bounds__(32)
void nms_seq_kernel(const unsigned* __restrict__ mask,
                    unsigned* __restrict__ rem_out) {
    __shared__ __align__(16) unsigned ring[RING_DEPTH * MASK_WORDS]; // 4KB
    __shared__ __align__(16) unsigned rem[MASK_WORDS];               // 512B
    int lane = threadIdx.x;

    reinterpret_cast<uint4*>(rem)[lane] = make_uint4(0u, 0u, 0u, 0u);

    unsigned ldsRing = (unsigned)(unsigned long long)(&ring[0]);
    unsigned long long gBase = (unsigned long long)mask;

    // Prologue: fill the ring with rows 0..7.
    #pragma unroll
    for (int r = 0; r < RING_DEPTH; ++r) {
        unsigned off = (unsigned)(r * 512 + lane * 16);
        unsigned lad = ldsRing + off;
        asm volatile("global_load_async_to_lds_b128 %0, %1, %2 offset:0"
                     :: "v"(lad), "v"(off), "s"(gBase) : "memory");
    }

    for (int i = 0; i < N_BOX - RING_DEPTH; ++i) {
        asm volatile("s_wait_asynccnt 0x7" ::: "memory");  // row i landed
        unsigned w = rem[i >> 5];                          // broadcast ds read
        if (!((w >> (i & 31)) & 1u)) {
            uint4 r  = reinterpret_cast<uint4*>(rem)[lane];
            uint4 rv = reinterpret_cast<uint4*>(&ring[(i & (RING_DEPTH - 1)) * MASK_WORDS])[lane];
            r.x |= rv.x; r.y |= rv.y; r.z |= rv.z; r.w |= rv.w;
            reinterpret_cast<uint4*>(rem)[lane] = r;
        }
        unsigned off = (unsigned)((i + RING_DEPTH) * 512 + lane * 16);
        unsigned lad = ldsRing + (unsigned)(((i + RING_DEPTH) & (RING_DEPTH - 1)) * 512)
                               + (unsigned)(lane * 16);
        asm volatile("global_load_async_to_lds_b128 %0, %1, %2 offset:0"
                     :: "v"(lad), "v"(off), "s"(gBase) : "memory");
    }
    asm volatile("s_wait_asynccnt 0x0" ::: "memory");      // drain tail
    for (int i = N_BOX - RING_DEPTH; i < N_BOX; ++i) {
        unsigned w = rem[i >> 5];
        if (!((w >> (i & 31)) & 1u)) {
            uint4 r  = reinterpret_cast<uint4*>(rem)[lane];
            uint4 rv = reinterpret_cast<uint4*>(&ring[(i & (RING_DEPTH - 1)) * MASK_WORDS])[lane];
            r.x |= rv.x; r.y |= rv.y; r.z |= rv.z; r.w |= rv.w;
            reinterpret_cast<uint4*>(rem)[lane] = r;
        }
    }
    reinterpret_cast<uint4*>(rem_out)[lane] = reinterpret_cast<uint4*>(rem)[lane];
}

// ---------------------------------------------------------------- kernel C3
__global__ __launch_bounds__(256)
void nms_scatter_kernel(const unsigned* __restrict__ rem,
                        const int*      __restrict__ sidx,
                        const float*    __restrict__ scores,
                        float*          __restrict__ out) {
    int s = blockIdx.x * 256 + threadIdx.x;
    int orig = sidx[s];
    bool keep = !((rem[s >> 5] >> (s & 31)) & 1u);
    out[orig] = keep ? scores[orig] : 0.0f;
}

// ---------------------------------------------------------------- fallback
// Fused single-WGP scan (used when workspace can't hold the 2MB mask).
__global__ __launch_bounds__(NTHREADS)
void nms_scan_fused_kernel(const float4* __restrict__ sboxes,
                           const int*    __restrict__ sidx,
                           const float*  __restrict__ scores,
                           float*        __restrict__ out) {
    __shared__ float4   sb[N_BOX];    // 64 KB of the 320 KB WGP LDS
    __shared__ unsigned rem[MASK_WORDS];
    int tid = threadIdx.x;
    if (tid < MASK_WORDS) rem[tid] = 0u;
    __builtin_prefetch(&sidx[tid * 4], 0, 3);
    __builtin_prefetch(&scores[tid * 4], 0, 3);
    {
        unsigned ldsBase = (unsigned)(unsigned long long)(&sb[0]);
        unsigned long long gBase = (unsigned long long)sboxes;
        #pragma unroll
        for (int c = 0; c < 4; ++c) {
            unsigned off = (unsigned)(c * (NTHREADS * 16) + tid * 16);
            unsigned lad = ldsBase + off;
            asm volatile("global_load_async_to_lds_b128 %0, %1, %2 offset:0"
                         :: "v"(lad), "v"(off), "s"(gBase) : "memory");
        }
        asm volatile("s_wait_asynccnt 0x0" ::: "memory");
    }
    __syncthreads();
    for (int i = 0; i < N_BOX; ++i) {
        unsigned w = rem[i >> 5];
        if (!((w >> (i & 31)) & 1u)) {
            float4 bi = sb[i];
            float  ai = box_area(bi);
            for (int j = i + 1 + tid; j < N_BOX; j += NTHREADS) {
                if (iou_gt(bi, ai, sb[j]))
                    atomicOr(&rem[j >> 5], 1u << (j & 31)); // ds_or_b32
            }
        }
        __syncthreads();
    }
    for (int s = tid; s < N_BOX; s += NTHREADS) {
        int orig = sidx[s];
        bool keep = !((rem[s >> 5] >> (s & 31)) & 1u);
        out[orig] = keep ? scores[orig] : 0.0f;
    }
}

// ---------------------------------------------------------------- launcher
extern "C" void kernel_launch(void* const* d_in, const int* in_sizes, int n_in,
                              void* d_out, int out_size, void* d_ws, size_t ws_size,
                              hipStream_t stream) {
    const float* boxes     = (const float*)d_in[0];   // (4096,4) f32
    const float* scores    = (const float*)d_in[1];   // (4096,)  f32
    const int*   class_ids = (const int*)d_in[2];     // (4096,)  i32
    float* out = (float*)d_out;                       // (4096,)  f32

    char* ws = (char*)d_ws;
    const size_t offMaxc   = 0;
    const size_t offSboxes = 256;
    const size_t offSidx   = offSboxes + (size_t)N_BOX * 16;       // +64KB
    const size_t offRem    = offSidx + (size_t)N_BOX * 4;          // +16KB
    const size_t offMask   = offRem + 512;
    const size_t needSplit = offMask + (size_t)N_BOX * MASK_WORDS * 4; // ~2.08MB

    float*    maxc   = (float*)(ws + offMaxc);
    float4*   sboxes = (float4*)(ws + offSboxes);
    int*      sidx   = (int*)(ws + offSidx);
    unsigned* remO   = (unsigned*)(ws + offRem);
    unsigned* mask   = (unsigned*)(ws + offMask);

    nms_maxred_kernel<<<1, 256, 0, stream>>>(boxes, maxc);
    nms_sort_kernel<<<1, NTHREADS, 0, stream>>>(scores, boxes, class_ids,
                                                maxc, sboxes, sidx);
    if (ws_size >= needSplit) {
        dim3 mg(N_BOX / 256, MASK_WORDS);       // (16, 128) blocks
        nms_mask_kernel<<<mg, 256, 0, stream>>>(sboxes, mask);
        nms_seq_kernel<<<1, 32, 0, stream>>>(mask, remO);
        nms_scatter_kernel<<<N_BOX / 256, 256, 0, stream>>>(remO, sidx, scores, out);
    } else {
        nms_scan_fused_kernel<<<1, NTHREADS, 0, stream>>>(sboxes, sidx, scores, out);
    }
}